// DCTTransform_51024211476909
// MI455X (gfx1250) — compile-verified
//
#include <hip/hip_runtime.h>

// 8x8 block DCT over (16,32,512,512) f32, via 16x16 block-diagonal DCT matrix
// and V_WMMA_F32_16X16X4_F32 on gfx1250 (wave32).
//
// Per 16x16 tile T_out = Db * T_in * Db^T, Db = diag(D8, D8).
// Memory-bound: 1 GiB total traffic -> ~46us floor @ 23.3 TB/s.

typedef __attribute__((ext_vector_type(2))) float v2f;
typedef __attribute__((ext_vector_type(8))) float v8f;

#define IMG_W      512
#define IMG_SHIFT  18          // 512*512 = 1<<18 elements per channel image
#define TILES_X    32          // 512/16
#define TILES_PER_IMG 1024     // 32*32
#define N_IMAGES   512         // 16*32
#define TOTAL_TILES (N_IMAGES * TILES_PER_IMG)  // 524288
#define WAVES_PER_BLOCK 8
#define BLOCKS 2048

// Db(i,j) for the 16x16 block-diagonal matrix built from the 8x8 DCT matrix d.
__device__ __forceinline__ float db_elem(const float* __restrict__ d, int i, int j) {
    return ((i >> 3) == (j >> 3)) ? d[((i & 7) << 3) + (j & 7)] : 0.0f;
}

__global__ __launch_bounds__(256) void dct8x8_wmma_kernel(
    const float* __restrict__ x,
    const float* __restrict__ dct,     // 8x8 row-major
    float* __restrict__ out)
{
    __shared__ float lds[WAVES_PER_BLOCK][256];   // one 16x16 f32 tile per wave

    const int lane = threadIdx.x & 31;
    const int wv   = threadIdx.x >> 5;
    const int nL   = lane & 15;       // M index (A-layout) == N index (B/C/D-layout)
    const int tL   = lane >> 4;       // half-wave selector -> K sub-pair

    float* tile = lds[wv];

    // ---- Hoist the Db operand registers (identical values serve as A of Db
    // ---- in stage 1 and as B of Db^T in stage 2, by lane-layout symmetry).
    v2f dbOp[4];
#pragma unroll
    for (int c = 0; c < 4; ++c) {
        int ka = 4 * c + 2 * tL;
        dbOp[c].x = db_elem(dct, nL, ka);
        dbOp[c].y = db_elem(dct, nL, ka + 1);
    }

    const int totalWaves = BLOCKS * WAVES_PER_BLOCK;
    const int waveId = blockIdx.x * WAVES_PER_BLOCK + wv;

    // Per-lane tile-load coordinates: lane covers row r, 8-float half hh.
    const int r  = lane >> 1;
    const int hh = (lane & 1) << 3;

    for (int t = waveId; t < TOTAL_TILES; t += totalWaves) {
        const int ch   = t >> 10;
        const int rem  = t & (TILES_PER_IMG - 1);
        const int row0 = (rem >> 5) << 4;
        const int col0 = (rem & (TILES_X - 1)) << 4;

        const float* src = x + (((size_t)ch) << IMG_SHIFT) + row0 * IMG_W + col0;
        float*       dst = out + (((size_t)ch) << IMG_SHIFT) + row0 * IMG_W + col0;

        // ---- Load 16x16 tile: 2x global_load_b128 per lane, fully coalesced.
        const float4* gp = (const float4*)(src + r * IMG_W + hh);
        float4 q0 = gp[0];
        float4 q1 = gp[1];

        // Prefetch next tile into L2 (global_prefetch_b8).
        int tn = t + totalWaves;
        if (tn < TOTAL_TILES) {
            int chn   = tn >> 10;
            int remn  = tn & (TILES_PER_IMG - 1);
            const float* nsrc = x + (((size_t)chn) << IMG_SHIFT)
                                  + (((remn >> 5) << 4) * IMG_W)
                                  + ((remn & (TILES_X - 1)) << 4);
            __builtin_prefetch(nsrc + r * IMG_W + hh, 0, 1);
        }

        // Stage tile row-major into this wave's private LDS slab.
        *(float4*)(tile + r * 16 + hh)     = q0;
        *(float4*)(tile + r * 16 + hh + 4) = q1;

        // ---- Stage 1: T = Db * X  (A = Db regs, B = X from LDS, B-layout).
        v8f t1 = {};
#pragma unroll
        for (int c = 0; c < 4; ++c) {
            int ka = 4 * c + 2 * tL;
            v2f bx;
            bx.x = tile[ka * 16 + nL];
            bx.y = tile[(ka + 1) * 16 + nL];
            t1 = __builtin_amdgcn_wmma_f32_16x16x4_f32(
                    false, dbOp[c], false, bx, (short)0, t1, false, false);
        }

        // ---- Spill T (C/D layout) to LDS row-major so it can be re-read in
        // ---- A-layout for stage 2. Same-wave LDS ops are in-order, so the
        // ---- stage-1 B reads above are safely ordered before these stores.
#pragma unroll
        for (int v = 0; v < 8; ++v) {
            tile[(v + 8 * tL) * 16 + nL] = t1[v];
        }

        // ---- Stage 2: Y = T * Db^T  (A = T from LDS, contiguous ds_load_b64;
        // ---- B = Db^T == dbOp by layout symmetry).
        v8f y = {};
#pragma unroll
        for (int c = 0; c < 4; ++c) {
            int ka = 4 * c + 2 * tL;
            v2f ax;
            ax.x = tile[nL * 16 + ka];
            ax.y = tile[nL * 16 + ka + 1];
            y = __builtin_amdgcn_wmma_f32_16x16x4_f32(
                    false, ax, false, dbOp[c], (short)0, y, false, false);
        }

        // ---- Store Y straight from C/D layout: each b32 store covers two
        // ---- contiguous 64B row segments across the wave -> coalesced.
#pragma unroll
        for (int v = 0; v < 8; ++v) {
            dst[(v + 8 * tL) * IMG_W + nL] = y[v];
        }
    }
}

extern "C" void kernel_launch(void* const* d_in, const int* in_sizes, int n_in,
                              void* d_out, int out_size, void* d_ws, size_t ws_size,
                              hipStream_t stream) {
    const float* x   = (const float*)d_in[0];
    const float* dct = (const float*)d_in[1];
    float* out       = (float*)d_out;
    (void)in_sizes; (void)n_in; (void)out_size; (void)d_ws; (void)ws_size;

    dct8x8_wmma_kernel<<<BLOCKS, 256, 0, stream>>>(x, dct, out);
}